// MPNCOV_64862596104493
// MI455X (gfx1250) — compile-verified
//
#include <hip/hip_runtime.h>
#include <math.h>

typedef __attribute__((ext_vector_type(2))) float v2f;
typedef __attribute__((ext_vector_type(8))) float v8f;

#define DD      256
#define MM      784          // 28*28
#define NB      128          // batch
#define BM      128
#define BN      128
#define KT      16
#define KP      (KT/2)       // k-pairs per LDS stage

// ---------------------------------------------------------------------------
// Column means: mu[b][j] = (1/M) sum_m x[b][m][j]
// ---------------------------------------------------------------------------
__global__ void mpncov_colsum(const float* __restrict__ x, float* __restrict__ mu) {
    int b = blockIdx.x;
    int j = threadIdx.x;                       // 0..255
    const float* xb = x + (size_t)b * MM * DD;
    float s = 0.f;
    for (int m = 0; m < MM; ++m) s += xb[(size_t)m * DD + j];
    mu[b * DD + j] = s * (1.0f / MM);
}

// ---------------------------------------------------------------------------
// prep: gram -> A (in place), Z0 = 0.5*(3I - A), normA = trace(cov)
// ---------------------------------------------------------------------------
__global__ void mpncov_prep(float* __restrict__ gram, const float* __restrict__ mu,
                            float* __restrict__ z0, float* __restrict__ norm_out) {
    __shared__ float red[256];
    int b = blockIdx.x;
    int t = threadIdx.x;
    float* g  = gram + (size_t)b * DD * DD;
    float* zb = z0   + (size_t)b * DD * DD;
    const float* m = mu + b * DD;
    float mt = m[t];

    red[t] = g[t * DD + t] * (1.0f / MM) - mt * mt;
    __syncthreads();
    for (int s = 128; s > 0; s >>= 1) {
        if (t < s) red[t] += red[t + s];
        __syncthreads();
    }
    float normA = red[0];
    if (t == 0) norm_out[b] = normA;
    float inv = 1.0f / normA;

    for (int i = 0; i < DD; ++i) {
        float a = (g[i * DD + t] * (1.0f / MM) - m[i] * mt) * inv;
        g[i * DD + t] = a;
        zb[i * DD + t] = 0.5f * ((i == t ? 3.0f : 0.0f) - a);
    }
}

// ---------------------------------------------------------------------------
// Batched f32 WMMA GEMM:  C[b] = epilogue( A[b] @ B[b] ), ld = 256 everywhere.
// TRANSA: A(i,k) read as Araw[k*256 + i] (Gram = X^T X).
// MODE 0: C = acc ; MODE 1: C = 0.5*(3I - acc)
//
// Block: 256 threads = 8 waves; block tile 128x128; wave tile 32x64
// (8 accumulators of v_wmma_f32_16x16x4_f32).
// LDS slabs are stored k-pair interleaved as float2 so every WMMA A/B
// fragment is a single aligned ds_load_b64 (no VGPR repacking).
//   As2[p][m] = { A(2p, m), A(2p+1, m) },  fragment pair index = kk/2 + (lane>=16)
// ---------------------------------------------------------------------------
template <bool TRANSA, int MODE>
__global__ void mpncov_gemm(const float* __restrict__ A, const float* __restrict__ B,
                            float* __restrict__ C, int K,
                            size_t strideA, size_t strideB, size_t strideC) {
    __shared__ float2 As2[KP][BM + 1];
    __shared__ float2 Bs2[KP][BN + 1];

    int b = blockIdx.y;
    const float* Ab = A + (size_t)b * strideA;
    const float* Bb = B + (size_t)b * strideB;
    float*       Cb = C + (size_t)b * strideC;

    int nb = blockIdx.x & 1;        // 2 n-blocks of 128
    int mb = blockIdx.x >> 1;       // 2 m-blocks of 128
    int i0 = mb * BM;
    int j0 = nb * BN;

    int tid  = threadIdx.x;
    int lane = tid & 31;
    int wave = tid >> 5;
    int wm   = wave >> 1;           // 0..3 -> row offset wm*32
    int wn   = wave & 1;            // 0..1 -> col offset wn*64

    v8f acc[2][4] = {};

    int mlane = lane & 15;
    int hi    = lane >> 4;          // pair select: lanes 16-31 take K+2,K+3

    float* asf = (float*)As2;       // flat views for component writes
    float* bsf = (float*)Bs2;

    for (int k0 = 0; k0 < K; k0 += KT) {
        if (TRANSA) {
            // scalar: m fastest -> coalesced reads of X rows
            for (int idx = tid; idx < KT * BM; idx += 256) {
                int k = idx >> 7, mi = idx & 127;
                asf[(k >> 1) * ((BM + 1) * 2) + mi * 2 + (k & 1)] =
                    Ab[(size_t)(k0 + k) * DD + i0 + mi];
            }
        } else {
            // row-major A: 8 float2 per row are contiguous -> vector loads
            for (int idx = tid; idx < KP * BM; idx += 256) {
                int mi = idx >> 3, p = idx & 7;  // 8 pairs per row
                const float2* ag =
                    (const float2*)(Ab + (size_t)(i0 + mi) * DD + k0 + 2 * p);
                As2[p][mi] = *ag;
            }
        }
        // B rows: n fastest -> coalesced; components land in pair slots
        for (int idx = tid; idx < KT * BN; idx += 256) {
            int k = idx >> 7, n = idx & 127;
            bsf[(k >> 1) * ((BN + 1) * 2) + n * 2 + (k & 1)] =
                Bb[(size_t)(k0 + k) * DD + j0 + n];
        }
        __syncthreads();

#pragma unroll
        for (int kk = 0; kk < KT; kk += 4) {
            int pidx = (kk >> 1) + hi;
            v2f afr[2], bfr[4];
#pragma unroll
            for (int ti = 0; ti < 2; ++ti) {
                float2 t = As2[pidx][wm * 32 + ti * 16 + mlane];
                afr[ti].x = t.x; afr[ti].y = t.y;
            }
#pragma unroll
            for (int tj = 0; tj < 4; ++tj) {
                float2 t = Bs2[pidx][wn * 64 + tj * 16 + mlane];
                bfr[tj].x = t.x; bfr[tj].y = t.y;
            }
#pragma unroll
            for (int ti = 0; ti < 2; ++ti)
#pragma unroll
                for (int tj = 0; tj < 4; ++tj)
                    acc[ti][tj] = __builtin_amdgcn_wmma_f32_16x16x4_f32(
                        /*neg_a=*/false, afr[ti], /*neg_b=*/false, bfr[tj],
                        /*c_mod=*/(short)0, acc[ti][tj],
                        /*reuse_a=*/false, /*reuse_b=*/false);
        }
        __syncthreads();
    }

    // Epilogue: C/D layout -> VGPR r holds rows r (lanes 0-15) / r+8 (16-31)
    int rowoff = hi << 3;
    int col    = mlane;
#pragma unroll
    for (int ti = 0; ti < 2; ++ti)
#pragma unroll
        for (int tj = 0; tj < 4; ++tj)
#pragma unroll
            for (int r = 0; r < 8; ++r) {
                int gi = i0 + wm * 32 + ti * 16 + rowoff + r;
                int gj = j0 + wn * 64 + tj * 16 + col;
                float v = acc[ti][tj][r];
                if (MODE == 1) v = 0.5f * ((gi == gj ? 3.0f : 0.0f) - v);
                Cb[(size_t)gi * DD + gj] = v;
            }
}

// ---------------------------------------------------------------------------
// triu extraction with sqrt(normA) scale
// ---------------------------------------------------------------------------
__global__ void mpncov_triu(const float* __restrict__ Y, const float* __restrict__ normA,
                            float* __restrict__ out) {
    int b = blockIdx.y;
    int i = blockIdx.x;
    int j = threadIdx.x;
    if (j < i) return;
    float s = sqrtf(normA[b]);
    size_t base = (size_t)b * (DD * (DD + 1) / 2);
    int idx = i * DD - (i * (i - 1)) / 2 + (j - i);
    out[base + idx] = Y[(size_t)b * DD * DD + (size_t)i * DD + j] * s;
}

// ---------------------------------------------------------------------------
extern "C" void kernel_launch(void* const* d_in, const int* in_sizes, int n_in,
                              void* d_out, int out_size, void* d_ws, size_t ws_size,
                              hipStream_t stream) {
    (void)in_sizes; (void)n_in; (void)out_size; (void)ws_size;
    const float* x = (const float*)d_in[0];
    float* out = (float*)d_out;
    float* ws  = (float*)d_ws;

    const size_t S = (size_t)NB * DD * DD;   // floats per batched 256x256 set
    float* W0 = ws;                          // gram -> A -> ZY
    float* W1 = ws + S;                      // Y ping
    float* W2 = ws + 2 * S;                  // Y pong
    float* W3 = ws + 3 * S;                  // Z ping
    float* W4 = ws + 4 * S;                  // Z pong
    float* mu    = ws + 5 * S;               // 128*256
    float* normA = mu + NB * DD;             // 128

    const size_t SX = (size_t)MM * DD;       // x batch stride
    const size_t SM = (size_t)DD * DD;
    dim3 gg(4, NB);                          // 4 tile-blocks per batch matrix

    // 1) means
    mpncov_colsum<<<dim3(NB), 256, 0, stream>>>(x, mu);
    // 2) gram = X^T X  (K = 784, transposed access into x)
    mpncov_gemm<true, 0><<<gg, 256, 0, stream>>>(x, x, W0, MM, SX, SX, SM);
    // 3) cov -> A (in W0), Z0 (W3), normA
    mpncov_prep<<<dim3(NB), 256, 0, stream>>>(W0, mu, W3, normA);
    // 4) Y0 = Z0 @ A
    mpncov_gemm<false, 0><<<gg, 256, 0, stream>>>(W3, W0, W1, DD, SM, SM, SM);

    float* Y  = W1; float* Yn = W2;
    float* Z  = W3; float* Zn = W4;
    float* ZY = W0;                           // A no longer needed

    for (int it = 1; it < 5; ++it) {
        mpncov_gemm<false, 1><<<gg, 256, 0, stream>>>(Z, Y, ZY, DD, SM, SM, SM);
        mpncov_gemm<false, 0><<<gg, 256, 0, stream>>>(Y, ZY, Yn, DD, SM, SM, SM);
        mpncov_gemm<false, 0><<<gg, 256, 0, stream>>>(ZY, Z, Zn, DD, SM, SM, SM);
        float* t;
        t = Y; Y = Yn; Yn = t;
        t = Z; Z = Zn; Zn = t;
    }

    mpncov_triu<<<dim3(DD, NB), 256, 0, stream>>>(Y, normA, out);
}